// KANRNNEncoder_53197464928834
// MI455X (gfx1250) — compile-verified
//
#include <hip/hip_runtime.h>

typedef float v8f __attribute__((ext_vector_type(8)));
typedef float v2f __attribute__((ext_vector_type(2)));

#define BB   1024   // batch
#define TT   1024   // timesteps
#define FF   8      // features
#define HH   256    // hidden
#define LL   64     // latent out

#define ROWS_PER_WAVE   16
#define WAVES_PER_BLOCK 2
#define ROWS_PER_BLOCK  (ROWS_PER_WAVE * WAVES_PER_BLOCK)   // 32
#define NTHREADS        (32 * WAVES_PER_BLOCK)              // 64 (2 x wave32)
#define SROW            260  // padded LDS row stride (floats); 260 % 64 == 4 -> conflict-free b64 reads

__device__ __forceinline__ float sigmoidf_(float x) {
    return 1.0f / (1.0f + __expf(-x));
}

__global__ __launch_bounds__(NTHREADS)
void kan_rnn_enc_kernel(const float* __restrict__ x_ctx,   // (B,T,F)
                        const float* __restrict__ a_in,    // (F,K) flat 64
                        const float* __restrict__ b_in,    // (F,K) flat 64
                        const float* __restrict__ a_h,     // (H,K) flat 2048
                        const float* __restrict__ b_h,     // (H,K) flat 2048
                        const float* __restrict__ W_lat,   // (L,H) row-major
                        const float* __restrict__ b_lat,   // (L,)
                        float* __restrict__ out) {         // (B,L)
    __shared__ float lx[ROWS_PER_BLOCK * FF];      // x at t = T-1
    __shared__ float px[ROWS_PER_BLOCK * FF];      // x at t = T-2
    __shared__ float sPrev[ROWS_PER_BLOCK * 24];   // h_{T-1}[:, 0:24]
    __shared__ float sH[ROWS_PER_BLOCK * SROW];    // hT, padded rows

    const int tid = threadIdx.x;
    const int r0  = blockIdx.x * ROWS_PER_BLOCK;

    // ---- Phase 1a: stage the only two timesteps of x that matter (contiguous 16 floats/row)
    for (int i = tid; i < ROWS_PER_BLOCK * FF; i += NTHREADS) {
        int row = i >> 3, f = i & 7;
        const float* base = x_ctx + (size_t)(r0 + row) * (TT * FF);
        px[i] = base[(TT - 2) * FF + f];
        lx[i] = base[(TT - 1) * FF + f];
    }
    __syncthreads();

    // ---- Phase 1b: the 24 surviving hidden units of h_{T-1}
    for (int i = tid; i < ROWS_PER_BLOCK * 24; i += NTHREADS) {
        int row = i / 24, hh = i % 24;
        int f = hh >> 3;
        float t = a_in[hh] * (px[row * FF + f] - b_in[hh]);
        sPrev[i] = sigmoidf_(2.0f * sigmoidf_(t));
    }
    __syncthreads();

    // ---- Phase 1c: final hidden state hT[row][0..255] into LDS
    for (int i = tid; i < ROWS_PER_BLOCK * HH; i += NTHREADS) {
        int row = i >> 8, j = i & 255;
        float v;
        if (j < 64) {
            int f = j >> 3;
            v = sigmoidf_(2.0f * sigmoidf_(a_in[j] * (lx[row * FF + f] - b_in[j])));
        } else {
            int m  = j - 64;        // flat (hh,k) index into a_h/b_h
            int hh = m >> 3;
            float ph = sPrev[row * 24 + hh];
            v = sigmoidf_(2.0f * sigmoidf_(a_h[m] * (ph - b_h[m])));
        }
        sH[row * SROW + j] = v;
    }
    __syncthreads();

    // ---- Phase 2: z = hT @ W_lat^T + b_lat via V_WMMA_F32_16X16X4_F32
    // Uniform control flow from here on: EXEC is all 1s (WMMA requirement).
    const int lane = tid & 31;
    const int wave = tid >> 5;
    const int nl   = lane & 15;          // N (and M) sub-index
    const int kb   = (lane >> 4) << 1;   // K pair base: 0 (lanes 0-15) or 2 (lanes 16-31)

    v8f acc[4];
#pragma unroll
    for (int n = 0; n < 4; ++n) {
        float bl = b_lat[n * 16 + nl];   // D lane holds N = lane%16 in every VGPR
#pragma unroll
        for (int r = 0; r < 8; ++r) acc[n][r] = bl;
    }

    // A fragment source: hT row (wave*16 + nl), K pair at kb (ISA 32-bit A 16x4 layout)
    const float* aRow  = &sH[(wave * 16 + nl) * SROW + kb];
    // B fragment source: W_lat[n*16+nl][k..k+1] (B row-of-K striped across lanes)
    const float* wBase = W_lat + kb;

#pragma unroll
    for (int kk = 0; kk < HH / 4; ++kk) {          // 64 K-chunks of 4
        v2f afrag = *(const v2f*)(aRow + kk * 4);  // ds_load_b64, bank-conflict-free
#pragma unroll
        for (int n = 0; n < 4; ++n) {
            v2f bfrag = *(const v2f*)(wBase + (n * 16 + nl) * HH + kk * 4); // global_load_b64
            acc[n] = __builtin_amdgcn_wmma_f32_16x16x4_f32(
                false, afrag, false, bfrag, (short)0, acc[n], false, false);
        }
    }

    // ---- Store D: VGPR r holds (M=r, N=lane) for lanes 0-15, (M=r+8, N=lane-16) for 16-31
    const int rowTop = r0 + wave * 16 + ((lane >> 4) << 3);
#pragma unroll
    for (int n = 0; n < 4; ++n) {
#pragma unroll
        for (int r = 0; r < 8; ++r) {
            out[(size_t)(rowTop + r) * LL + n * 16 + nl] = acc[n][r];
        }
    }
}

extern "C" void kernel_launch(void* const* d_in, const int* in_sizes, int n_in,
                              void* d_out, int out_size, void* d_ws, size_t ws_size,
                              hipStream_t stream) {
    (void)in_sizes; (void)n_in; (void)out_size; (void)d_ws; (void)ws_size;
    const float* x_ctx = (const float*)d_in[0];
    const float* a_in  = (const float*)d_in[1];
    const float* b_in  = (const float*)d_in[2];
    const float* a_h   = (const float*)d_in[3];
    const float* b_h   = (const float*)d_in[4];
    const float* W_lat = (const float*)d_in[5];
    const float* b_lat = (const float*)d_in[6];
    float* out = (float*)d_out;

    dim3 grid(BB / ROWS_PER_BLOCK);   // 32 blocks
    dim3 block(NTHREADS);             // 64 threads = 2 waves
    hipLaunchKernelGGL(kan_rnn_enc_kernel, grid, block, 0, stream,
                       x_ctx, a_in, b_in, a_h, b_h, W_lat, b_lat, out);
}